// Node2GraphLayer_35631048688006
// MI455X (gfx1250) — compile-verified
//
#include <hip/hip_runtime.h>
#include <math.h>

// Problem constants (from reference): X [32, 8192, 256] f32, W [512, 1024], b [512]
#define BATCH   32
#define NODES   8192
#define DIM     256
#define SPLITS  32
#define ROWS_PER_SPLIT (NODES / SPLITS)   // 256 rows per block
#define KDIM    1024                      // 4*DIM
#define EMBG    512

typedef float v2f __attribute__((ext_vector_type(2)));
typedef float v4f __attribute__((ext_vector_type(4)));
typedef float v8f __attribute__((ext_vector_type(8)));

// ---------------------------------------------------------------------------
// Kernel 1: streaming partial reduction over nodes.
// Block = (batch b, split s). 256 threads = 4 row-groups x 64 lanes.
// Lane l of group g loads a float4 covering dims [4l, 4l+3] of row n
// -> global_load_b128 th:TH_LOAD_NT (X is single-use, 268 MB > 192 MB L2:
// stream it non-temporally so L2 stays free for the reused partials/W/z).
// 64 lanes * 16B = 1 KB fully coalesced per row per group.
// ---------------------------------------------------------------------------
__global__ __launch_bounds__(256) void n2g_partial(
    const float* __restrict__ X,
    float* __restrict__ sumP, float* __restrict__ sqP,
    float* __restrict__ minP, float* __restrict__ maxP)
{
    const int s = blockIdx.x;            // node split
    const int b = blockIdx.y;            // batch
    const int g = threadIdx.x >> 6;      // row group 0..3
    const int l = threadIdx.x & 63;      // dim quad 0..63 (dims 4l..4l+3)

    const v4f* __restrict__ Xrow =
        (const v4f*)(X + ((size_t)b * NODES + (size_t)s * ROWS_PER_SPLIT) * DIM);

    const float INF = __builtin_inff();
    v4f s4 = {0.f, 0.f, 0.f, 0.f};
    v4f q4 = {0.f, 0.f, 0.f, 0.f};
    v4f mn = { INF,  INF,  INF,  INF};
    v4f mx = {-INF, -INF, -INF, -INF};

    #pragma unroll 4
    for (int n = g; n < ROWS_PER_SPLIT; n += 4) {
        v4f v = __builtin_nontemporal_load(&Xrow[(size_t)n * (DIM / 4) + l]);
        #pragma unroll
        for (int c = 0; c < 4; ++c) {
            s4[c] += v[c];
            q4[c]  = fmaf(v[c], v[c], q4[c]);
            mn[c]  = fminf(mn[c], v[c]);
            mx[c]  = fmaxf(mx[c], v[c]);
        }
    }

    __shared__ v4f shS[256], shQ[256], shMn[256], shMx[256];  // 16 KB LDS
    shS[threadIdx.x] = s4; shQ[threadIdx.x] = q4;
    shMn[threadIdx.x] = mn; shMx[threadIdx.x] = mx;
    __syncthreads();

    if (g == 0) {
        #pragma unroll
        for (int gg = 1; gg < 4; ++gg) {
            int t = gg * 64 + l;
            v4f a0 = shS[t], a1 = shQ[t], a2 = shMn[t], a3 = shMx[t];
            #pragma unroll
            for (int c = 0; c < 4; ++c) {
                s4[c] += a0[c];
                q4[c] += a1[c];
                mn[c]  = fminf(mn[c], a2[c]);
                mx[c]  = fmaxf(mx[c], a3[c]);
            }
        }
        const size_t base = ((size_t)b * SPLITS + s) * (DIM / 4) + l;
        ((v4f*)sumP)[base] = s4;    // partials stay Regular-Temporal: kernel 2
        ((v4f*)sqP )[base] = q4;    // re-reads them, 4 MB fits easily in L2
        ((v4f*)minP)[base] = mn;
        ((v4f*)maxP)[base] = mx;
    }
}

// ---------------------------------------------------------------------------
// Kernel 2: combine SPLITS partials per (b,d); emit z[b, 0:256]=mean,
// [256:512]=min, [512:768]=max, [768:1024]=std (ddof=1), matching the concat.
// ---------------------------------------------------------------------------
__global__ __launch_bounds__(256) void n2g_finalize(
    const float* __restrict__ sumP, const float* __restrict__ sqP,
    const float* __restrict__ minP, const float* __restrict__ maxP,
    float* __restrict__ z)
{
    const int b = blockIdx.x;
    const int d = threadIdx.x;
    const float INF = __builtin_inff();
    float s = 0.f, q = 0.f, mn = INF, mx = -INF;
    #pragma unroll 4
    for (int sp = 0; sp < SPLITS; ++sp) {
        const size_t idx = ((size_t)b * SPLITS + sp) * DIM + d;
        s += sumP[idx];
        q += sqP[idx];
        mn = fminf(mn, minP[idx]);
        mx = fmaxf(mx, maxP[idx]);
    }
    const float invN  = 1.0f / (float)NODES;
    const float mean  = s * invN;
    float var = (q - s * s * invN) * (1.0f / (float)(NODES - 1));
    var = fmaxf(var, 0.0f);
    float* zb = z + (size_t)b * KDIM;
    zb[d]           = mean;
    zb[d + DIM]     = mn;
    zb[d + 2 * DIM] = mx;
    zb[d + 3 * DIM] = sqrtf(var);
}

// ---------------------------------------------------------------------------
// Kernel 3: out[32,512] = z[32,1024] @ W^T + bias via v_wmma_f32_16x16x4_f32.
// One wave (32 lanes) per 16x16 output tile; 2 M-tiles x 32 N-tiles = 64 waves.
// A (16x4 f32): lane m = lane&15, VGPR v holds K = (lane>>4)*2 + v.
// B (4x16 f32): lane n = lane&15, VGPR v holds K = (lane>>4)*2 + v  (B[k][n]=W[n][k]).
// C/D (16x16 f32, 8 VGPRs): VGPR r -> M = r + (lane>>4)*8, N = lane&15.
// fp32 WMMA keeps full reference precision; the GEMM is off the critical path.
// ---------------------------------------------------------------------------
__global__ __launch_bounds__(32) void n2g_gemm(
    const float* __restrict__ z, const float* __restrict__ W,
    const float* __restrict__ bias, float* __restrict__ out)
{
    const int wave = blockIdx.x;
    const int mt = wave & 1;        // 0..1  (M = 32)
    const int nt = wave >> 1;       // 0..31 (N = 512)
    const int lane = threadIdx.x;   // blockDim.x == 32, EXEC all-ones

    const int m  = mt * 16 + (lane & 15);
    const int n  = nt * 16 + (lane & 15);
    const int kb = (lane >> 4) * 2; // K base within the step for this half-wave

    const float* __restrict__ zrow = z + (size_t)m * KDIM + kb;
    const float* __restrict__ wrow = W + (size_t)n * KDIM + kb;

    v8f c = {};
    #pragma unroll 8
    for (int kk = 0; kk < KDIM; kk += 4) {
        v2f a = *(const v2f*)(zrow + kk);   // 8B-aligned (kb is 0 or 2)
        v2f bfr = *(const v2f*)(wrow + kk);
        c = __builtin_amdgcn_wmma_f32_16x16x4_f32(
                /*neg_a=*/false, a, /*neg_b=*/false, bfr,
                /*c_mod=*/(short)0, c, /*reuse_a=*/false, /*reuse_b=*/false);
    }

    const int rowM = mt * 16 + (lane >> 4) * 8;
    const int colN = n;
    const float bv = bias[colN];
    #pragma unroll
    for (int r = 0; r < 8; ++r)
        out[(size_t)(rowM + r) * EMBG + colN] = c[r] + bv;
}

// ---------------------------------------------------------------------------
// Host-side launch. Workspace layout (floats):
//   [0, P)   sumP   [P, 2P) sqP   [2P, 3P) minP   [3P, 4P) maxP
//   [4P, 4P + 32*1024)  z          where P = 32*32*256 = 262144  (~4.2 MB total)
// ---------------------------------------------------------------------------
extern "C" void kernel_launch(void* const* d_in, const int* in_sizes, int n_in,
                              void* d_out, int out_size, void* d_ws, size_t ws_size,
                              hipStream_t stream) {
    (void)in_sizes; (void)n_in; (void)out_size; (void)ws_size;
    const float* X    = (const float*)d_in[0];
    const float* W    = (const float*)d_in[1];
    const float* bias = (const float*)d_in[2];
    float* out = (float*)d_out;
    float* ws  = (float*)d_ws;

    const size_t P = (size_t)BATCH * SPLITS * DIM;   // 262144 floats per stat
    float* sumP = ws;
    float* sqP  = ws + P;
    float* minP = ws + 2 * P;
    float* maxP = ws + 3 * P;
    float* z    = ws + 4 * P;                        // 32*1024 floats

    n2g_partial <<<dim3(SPLITS, BATCH), 256, 0, stream>>>(X, sumP, sqP, minP, maxP);
    n2g_finalize<<<BATCH, DIM, 0, stream>>>(sumP, sqP, minP, maxP, z);
    n2g_gemm    <<<BATCH * EMBG / 256, 32, 0, stream>>>(z, W, bias, out);
}